// VanillaLI_71889162600692
// MI455X (gfx1250) — compile-verified
//
#include <hip/hip_runtime.h>
#include <stdint.h>

// Problem constants (reference: T=512, B=64, N=2048, f32)
#define T_TOT    512
#define CCH      131072                 // B*N independent channels
#define ROWB     (CCH * 4)              // bytes per time row = 524288
#define CH_T     8                      // timesteps staged per chunk
#define NCHUNK   (T_TOT / CH_T)         // 64 chunks
#define THREADS  256
#define CH_PER_T 2                      // channels per thread (float2)
#define WG_CH    (THREADS * CH_PER_T)   // 512 channels per workgroup
#define NBLOCKS  (CCH / WG_CH)          // 256 workgroups
#define BUF_FLTS (CH_T * WG_CH)         // 4096 floats = 16 KB per buffer
#define BUF_BYTES (BUF_FLTS * 4)
#define DEPTH    3                      // pipeline depth: 3 chunks in flight

typedef __attribute__((ext_vector_type(2))) float v2f;

__global__ __launch_bounds__(THREADS) void li_scan_async(
    const float* __restrict__ x, const float* __restrict__ betap,
    float* __restrict__ out) {
  // Triple-buffered staging area: 3 x 16 KB = 48 KB LDS per workgroup
  // (320 KB/WGP, so several workgroups still co-reside per WGP).
  __shared__ float buf[DEPTH * BUF_FLTS];

  const int      lidx    = threadIdx.x;
  const int      c0      = blockIdx.x * WG_CH + lidx * CH_PER_T;
  const uint32_t ch_byte = (uint32_t)c0 * 4u;  // lane's byte offset in a time row

  // LDS byte address of this lane's slot (low 32 bits of the flat shared
  // address are the LDS offset on gfx1250 per the aperture mapping).
  const uint32_t lds_lane =
      (uint32_t)(uintptr_t)(&buf[0]) + (uint32_t)lidx * 8u;

  // Issue the 8 async global->LDS b64 loads (one per timestep in the chunk)
  // for chunk k into buffer (k % DEPTH). Tracked with ASYNCcnt, not VGPRs.
  // Input is streamed exactly once (256 MiB > 192 MB L2) -> non-temporal.
  auto issue = [&](int k) {
    const char*    saddr = (const char*)x + (size_t)k * CH_T * (size_t)ROWB;
    const uint32_t lbase = lds_lane + (uint32_t)(k % DEPTH) * (uint32_t)BUF_BYTES;
#pragma unroll
    for (int dt = 0; dt < CH_T; ++dt) {
      const uint32_t vofs  = ch_byte + (uint32_t)dt * (uint32_t)ROWB;
      const uint32_t laddr = lbase + (uint32_t)dt * (uint32_t)(WG_CH * 4);
      asm volatile("global_load_async_to_lds_b64 %0, %1, %2 th:TH_LOAD_NT"
                   :
                   : "v"(laddr), "v"(vofs), "s"(saddr)
                   : "memory");
    }
  };

  const float beta = betap[0];
  v2f y;
  y.x = 0.0f;
  y.y = 0.0f;

  // Prologue: DEPTH chunks in flight = 24 outstanding async b64 loads/wave.
  // Across 2048 waves that is ~12 MB in flight, sized to the HBM
  // latency-bandwidth product at 23.3 TB/s.
  issue(0);
  issue(1);
  issue(2);

  for (int k = 0; k < NCHUNK; ++k) {
    // Async loads retire in order: chunk k is complete once ASYNCcnt drops
    // to 8 * (#newer chunks still outstanding).
    const int newer = NCHUNK - 1 - k;
    if (newer >= 2) {
      asm volatile("s_wait_asynccnt 0x10" ::: "memory");
    } else if (newer == 1) {
      asm volatile("s_wait_asynccnt 0x8" ::: "memory");
    } else {
      asm volatile("s_wait_asynccnt 0x0" ::: "memory");
    }

    const float* src   = &buf[(k % DEPTH) * BUF_FLTS];
    const size_t obase = (size_t)k * CH_T * CCH + (size_t)c0;

#pragma unroll
    for (int t = 0; t < CH_T; ++t) {
      // ds_load_b64 of this lane's two channels for timestep t.
      const v2f xv = *(const v2f*)(src + t * WG_CH + lidx * CH_PER_T);
      y = beta * y + xv;  // sequential recurrence carry stays in VGPRs
      // Write-once output: non-temporal b64 store, perfectly coalesced.
      __builtin_nontemporal_store(y, (v2f*)(out + obase + (size_t)t * CCH));
    }

    // Refill the buffer we just drained (its values were already consumed
    // into the FMA chain, which forced the dscnt wait before this point).
    if (k + DEPTH < NCHUNK) issue(k + DEPTH);
  }
}

extern "C" void kernel_launch(void* const* d_in, const int* in_sizes, int n_in,
                              void* d_out, int out_size, void* d_ws,
                              size_t ws_size, hipStream_t stream) {
  const float* x    = (const float*)d_in[0];
  const float* beta = (const float*)d_in[1];
  float*       out  = (float*)d_out;
  (void)in_sizes; (void)n_in; (void)out_size; (void)d_ws; (void)ws_size;

  li_scan_async<<<dim3(NBLOCKS), dim3(THREADS), 0, stream>>>(x, beta, out);
}